// max_layer_11020886081952
// MI455X (gfx1250) — compile-verified
//
#include <hip/hip_runtime.h>

// Problem constants (reference: N=16, A=4096, B=1024, fp32).
#define NB 16
#define AD 4096
#define BD 1024

typedef float f32x4 __attribute__((ext_vector_type(4)));

// ---------------------------------------------------------------------------
// Kernel 1: per-row argmax over B=1024, set flags[n*BD + idx] = 1.
// One wave32 per row (8 waves / 256-thread block). Each lane reads 8 f32x4s
// (coalesced: 32 lanes * 16B = 512B contiguous per step).
//
// L2 management (192 MB global L2, X = 256 MB): rows r >= BD can never be
// re-read by pass 2 (argmax indices are < B), so load them non-temporally
// (th:NT); rows r < BD (64 MB total) use regular-temporal loads so they stay
// L2-resident for the pass-2 copy. The branch is wave-uniform.
// NOTE: the NT arm's pointer is laundered through an empty asm so GVN cannot
// prove the two arms load identical addresses and merge them (round-2 codegen
// dropped the !nontemporal metadata exactly that way).
//
// First-occurrence argmax semantics (jnp.argmax): per-lane indices are
// visited in ascending order with strictly-greater updates; the cross-lane
// reduction breaks value ties by the smaller index.
// ---------------------------------------------------------------------------
__global__ __launch_bounds__(256) void argmax_flags_kernel(
    const float* __restrict__ X, unsigned char* __restrict__ flags) {
  const int wave = threadIdx.x >> 5;   // 0..7
  const int lane = threadIdx.x & 31;   // wave32 lane
  const long long row = (long long)blockIdx.x * 8 + wave;  // 0 .. N*A-1
  const int r = (int)(row & (AD - 1));                     // row % AD
  const f32x4* __restrict__ rp4 =
      reinterpret_cast<const f32x4*>(X + row * (long long)BD);

  f32x4 v[8];
  if (r >= BD) {
    // Streamed once, never re-read: non-temporal (th:NT) loads.
    // Opaque pointer copy: keeps this arm's loads distinct from the RT arm
    // so the nontemporal metadata survives GVN/hoisting.
    const f32x4* p = rp4;
    asm volatile("" : "+v"(p));
#pragma unroll
    for (int i = 0; i < 8; ++i)
      v[i] = __builtin_nontemporal_load(p + (i * 32 + lane));
  } else {
    // Candidate for pass-2 re-read: regular temporal loads (keep in L2).
#pragma unroll
    for (int i = 0; i < 8; ++i) v[i] = rp4[i * 32 + lane];
  }

  float maxv = -__builtin_inff();
  int maxb = 0;
#pragma unroll
  for (int i = 0; i < 8; ++i) {
    const int e = (i * 32 + lane) * 4;  // float index of this lane's f32x4
    if (v[i].x > maxv) { maxv = v[i].x; maxb = e + 0; }
    if (v[i].y > maxv) { maxv = v[i].y; maxb = e + 1; }
    if (v[i].z > maxv) { maxv = v[i].z; maxb = e + 2; }
    if (v[i].w > maxv) { maxv = v[i].w; maxb = e + 3; }
  }

  // wave32 xor-shuffle reduction; keep larger value, smaller index on ties.
#pragma unroll
  for (int off = 16; off >= 1; off >>= 1) {
    const float ov = __shfl_xor(maxv, off, 32);
    const int ob = __shfl_xor(maxb, off, 32);
    if (ov > maxv || (ov == maxv && ob < maxb)) {
      maxv = ov;
      maxb = ob;
    }
  }

  if (lane == 0) {
    const int n = (int)(row >> 12);           // row / AD
    flags[n * BD + maxb] = (unsigned char)1;  // racing writers all write 1
  }
}

// ---------------------------------------------------------------------------
// Kernel 2: one 256-thread block per output row (n, r). Each thread handles
// one f32x4 (256 * 4 = 1024 floats). The flag test is block-uniform:
// flagged rows stream X -> out (the X read should hit L2 thanks to pass-1
// temporal policy); unflagged rows (including all r >= 1024) store the
// 1e-15 constant without reading X — saves ~75% of pass-2 read bandwidth.
// All output stores are non-temporal: 256 MB written once, never re-read,
// so it must not evict the X rows we want resident in L2.
// ---------------------------------------------------------------------------
__global__ __launch_bounds__(256) void scatter_rows_kernel(
    const float* __restrict__ X, const unsigned char* __restrict__ flags,
    float* __restrict__ out) {
  const long long row = blockIdx.x;       // 0 .. N*A-1
  const int n = (int)(row >> 12);         // row / AD
  const int r = (int)(row & (AD - 1));    // row % AD
  const long long base = row * (long long)BD;

  const bool keep = (r < BD) && (flags[n * BD + r] != 0);

  f32x4* __restrict__ o4 = reinterpret_cast<f32x4*>(out + base) + threadIdx.x;
  if (keep) {
    const f32x4 val = reinterpret_cast<const f32x4*>(X + base)[threadIdx.x];
    __builtin_nontemporal_store(val, o4);
  } else {
    const f32x4 fill = {1e-15f, 1e-15f, 1e-15f, 1e-15f};
    __builtin_nontemporal_store(fill, o4);
  }
}

// ---------------------------------------------------------------------------
// Launch: memset flags (16 KB in d_ws) -> argmax pass -> scatter pass.
// All on `stream`; hipMemsetAsync is graph-capturable.
// ---------------------------------------------------------------------------
extern "C" void kernel_launch(void* const* d_in, const int* in_sizes, int n_in,
                              void* d_out, int out_size, void* d_ws,
                              size_t ws_size, hipStream_t stream) {
  const float* X = (const float*)d_in[0];
  float* out = (float*)d_out;
  unsigned char* flags = (unsigned char*)d_ws;  // NB*BD = 16384 bytes

  hipMemsetAsync(flags, 0, NB * BD, stream);

  // 16*4096 rows, 8 rows (waves) per block.
  argmax_flags_kernel<<<(NB * AD) / 8, 256, 0, stream>>>(X, flags);

  // One block per output row.
  scatter_rows_kernel<<<NB * AD, 256, 0, stream>>>(X, flags, out);
}